// Distance2Centroids_31705448579703
// MI455X (gfx1250) — compile-verified
//
#include <hip/hip_runtime.h>

typedef __attribute__((ext_vector_type(2))) float v2f;
typedef __attribute__((ext_vector_type(8))) float v8f;
typedef __attribute__((ext_vector_type(4))) unsigned int u32x4;
typedef __attribute__((ext_vector_type(4))) int i32x4;
typedef __attribute__((ext_vector_type(8))) int i32x8;

#define D_DIM   256
#define N_CENT  4096
#define M_ROWS  16384
#define BK      64
#define LDW     68                      // LDS row stride (floats): 64 + 4 TDM pad dwords
#define PANEL_FLOATS (128 * LDW)        // one 128 x 64 panel, padded
#define STAGE_FLOATS (2 * PANEL_FLOATS) // A panel + B panel
#define PANEL_BYTES  (PANEL_FLOATS * 4)
#define STAGE_BYTES  (STAGE_FLOATS * 4)
#define LDS_BYTES    (2 * STAGE_BYTES)  // double buffered

// ---------------------------------------------------------------------------
// TDM: DMA a 128-row x 64-float tile (row stride 256 floats in memory) into
// LDS, padding +4 dwords after every 64 dwords (LDS row stride becomes 68).
// Descriptor layout per CDNA5 ISA ch.8 (D# groups 0/1; groups 2/3 zero = 2D).
// ---------------------------------------------------------------------------
__device__ __forceinline__ void tdm_load_panel(const float* gptr, unsigned lds_addr) {
  unsigned long long ga = (unsigned long long)gptr;
  u32x4 g0;
  g0[0] = 1u;                                               // count=1, no gather
  g0[1] = lds_addr;                                         // LDS byte address
  g0[2] = (unsigned)ga;                                     // global_addr[31:0]
  g0[3] = ((unsigned)(ga >> 32) & 0x1FFFFFFu) | (2u << 30); // global_addr[56:32], type=2
  i32x8 g1;
  g1[0] = (2 << 16) | (1 << 20) | (5 << 22) | (3 << 25);    // dsize=4B, pad_en, every 64dw, +4dw
  g1[1] = (int)(64u << 16);                                 // tensor_dim0 = 64   [79:48]
  g1[2] = (int)(128u << 16);                                // tensor_dim1 = 128  [111:80]
  g1[3] = (int)(64u << 16);                                 // tile_dim0 = 64     [127:112]
  g1[4] = 128;                                              // tile_dim1 = 128, tile_dim2 = 0
  g1[5] = 256;                                              // tensor_dim0_stride = 256
  g1[6] = 0;
  g1[7] = 0;
  i32x4 z4 = {};
#if defined(__clang_major__) && (__clang_major__ >= 23)
  i32x8 z8 = {};
  __builtin_amdgcn_tensor_load_to_lds(g0, g1, z4, z4, z8, 0);
#else
  __builtin_amdgcn_tensor_load_to_lds(g0, g1, z4, z4, 0);
#endif
}

// ---------------------------------------------------------------------------
// Kernel 1: squared L2 norms of every x row (16384) and every centroid (4096).
// ---------------------------------------------------------------------------
__global__ __launch_bounds__(256) void poincare_norms(const float* __restrict__ x,
                                                      const float* __restrict__ cent,
                                                      float* __restrict__ xn,
                                                      float* __restrict__ cn) {
  int row  = blockIdx.x * 8 + (threadIdx.x >> 5);
  int lane = threadIdx.x & 31;
  const float* src;
  float* dst;
  if (row < M_ROWS) {
    src = x + (size_t)row * D_DIM;
    dst = xn + row;
  } else {
    int r = row - M_ROWS;
    src = cent + (size_t)r * D_DIM;
    dst = cn + r;
  }
  float s = 0.f;
#pragma unroll
  for (int i = 0; i < D_DIM / 32; ++i) {
    float v = src[lane + i * 32];
    s = fmaf(v, v, s);
  }
#pragma unroll
  for (int off = 16; off > 0; off >>= 1) s += __shfl_xor(s, off, 32);
  if (lane == 0) *dst = s;
}

// ---------------------------------------------------------------------------
// Kernel 2: WMMA fp32 GEMM with TDM double-buffered LDS staging and fused
// Poincare distance epilogue. 128x128 tile per 256-thread block (8 waves),
// each wave = 2x4 tiles of v_wmma_f32_16x16x4_f32.
// ---------------------------------------------------------------------------
__global__ __launch_bounds__(256) void poincare_dist2(const float* __restrict__ x,
                                                      const float* __restrict__ cent,
                                                      const float* __restrict__ xn,
                                                      const float* __restrict__ cn,
                                                      float* __restrict__ out) {
  extern __shared__ float lds[];

  const int tid    = threadIdx.x;
  const int lane   = tid & 31;
  const int wid    = tid >> 5;
  const int lane15 = lane & 15;
  const int hi     = lane >> 4;        // half-wave select
  const int k2     = hi * 2;           // K offset of this half-wave in frags
  const int waveM  = (wid & 3) * 32;   // 4 wave-rows  x 32
  const int waveN  = (wid >> 2) * 64;  // 2 wave-cols  x 64
  const int blockM = blockIdx.y * 128;
  const int blockN = blockIdx.x * 128;

  const unsigned ldsBase = (unsigned)(unsigned long long)(void*)lds; // LDS byte addr

  // prologue: DMA chunk 0 into stage 0
  if (wid == 0) {
    tdm_load_panel(x    + (size_t)blockM * D_DIM, ldsBase);
    tdm_load_panel(cent + (size_t)blockN * D_DIM, ldsBase + PANEL_BYTES);
  }

  v8f acc[2][4];
#pragma unroll
  for (int im = 0; im < 2; ++im)
#pragma unroll
    for (int in = 0; in < 4; ++in)
      acc[im][in] = {};

#pragma unroll
  for (int c = 0; c < D_DIM / BK; ++c) {           // 4 chunks of K=64
    if (wid == 0) {
      if (c < D_DIM / BK - 1) {                    // prefetch next chunk
        const int kb = (c + 1) * BK;
        const unsigned so = ((unsigned)((c + 1) & 1)) * STAGE_BYTES;
        tdm_load_panel(x    + (size_t)blockM * D_DIM + kb, ldsBase + so);
        tdm_load_panel(cent + (size_t)blockN * D_DIM + kb, ldsBase + so + PANEL_BYTES);
      }
      if (c < D_DIM / BK - 1)
        __builtin_amdgcn_s_wait_tensorcnt(2);      // current chunk landed, next in flight
      else
        __builtin_amdgcn_s_wait_tensorcnt(0);
    }
    __syncthreads();

    const float* sA = lds + (c & 1) * STAGE_FLOATS;
    const float* sB = sA + PANEL_FLOATS;

#pragma unroll
    for (int kk = 0; kk < BK; kk += 4) {           // 16 k-steps x 8 WMMA
      v2f af[2], bf[4];
#pragma unroll
      for (int im = 0; im < 2; ++im)
        af[im] = *(const v2f*)(&sA[(waveM + im * 16 + lane15) * LDW + kk + k2]);
#pragma unroll
      for (int in = 0; in < 4; ++in)
        bf[in] = *(const v2f*)(&sB[(waveN + in * 16 + lane15) * LDW + kk + k2]);
#pragma unroll
      for (int im = 0; im < 2; ++im)
#pragma unroll
        for (int in = 0; in < 4; ++in)
          acc[im][in] = __builtin_amdgcn_wmma_f32_16x16x4_f32(
              false, af[im], false, bf[in], (short)0, acc[im][in], false, false);
    }
    __syncthreads();                               // stage free before reuse
  }

  // --- fused Poincare epilogue ---
  // alpha = 1 - 2d + b ; beta = 1 - a ; gamma = 1 - 2d + a*b
  // sq    = (alpha^2 a - 2 alpha beta d + beta^2 b) / gamma^2
  // dist^2 = ln((1+t)/(1-t))^2 with t = min(sqrt(sq), 1-1e-5)
  const float LN2SQ = 0.4804530139182014f;  // ln(2)^2
#pragma unroll
  for (int in = 0; in < 4; ++in) {
    int n = blockN + waveN + in * 16 + lane15;
    float a     = cn[n];
    float beta  = 1.f - a;
    float beta2 = beta * beta;
#pragma unroll
    for (int im = 0; im < 2; ++im) {
      v8f av = acc[im][in];
#pragma unroll
      for (int i = 0; i < 8; ++i) {
        int   m     = blockM + waveM + im * 16 + hi * 8 + i;
        float b     = xn[m];
        float d     = av[i];
        float alpha = fmaf(-2.f, d, 1.f + b);
        float gamma = fmaf(-2.f, d, fmaf(a, b, 1.f));
        float u     = alpha * a;
        float v     = beta * d;
        float w     = fmaf(-2.f, v, u);                 // alpha*a - 2*beta*d
        float num   = fmaf(alpha, w, beta2 * b);
        float rg    = __builtin_amdgcn_rcpf(gamma);
        float nrm   = __builtin_amdgcn_sqrtf(fmaxf(num, 0.f)) * fabsf(rg);
        float t     = fminf(nrm, 1.f - 1e-5f);
        float ratio = (1.f + t) * __builtin_amdgcn_rcpf(1.f - t);
        float l2    = __log2f(ratio);
        float res   = l2 * l2 * LN2SQ;                  // dist^2
        __builtin_nontemporal_store(res, out + (size_t)m * N_CENT + n);
      }
    }
  }
}

// ---------------------------------------------------------------------------
extern "C" void kernel_launch(void* const* d_in, const int* in_sizes, int n_in,
                              void* d_out, int out_size, void* d_ws, size_t ws_size,
                              hipStream_t stream) {
  const float* x    = (const float*)d_in[0];   // [8,2048,256]
  const float* cent = (const float*)d_in[1];   // [4096,256]
  float* out = (float*)d_out;                  // [8,2048,4096]
  float* xn  = (float*)d_ws;                   // 16384 floats
  float* cn  = xn + M_ROWS;                    // 4096 floats

  poincare_norms<<<(M_ROWS + N_CENT) / 8, 256, 0, stream>>>(x, cent, xn, cn);

  dim3 grid(N_CENT / 128, M_ROWS / 128);       // (32, 128)
  poincare_dist2<<<grid, 256, LDS_BYTES, stream>>>(x, cent, xn, cn, out);
}